// DiagLinear_88605175316958
// MI455X (gfx1250) — compile-verified
//
#include <hip/hip_runtime.h>

// DiagLinear: out = x * w + b  (per-feature scale + shift)
// x: [16384, 2048] f32, w/b: [2048] f32, out: [16384, 2048] f32
// Bandwidth-bound stream (~268 MB @ 23.3 TB/s => ~11.5 us floor).
// WMMA is useless here (0.25 FLOP/byte); the CDNA5 win is the async
// global->LDS b128 pipeline (ASYNCcnt) + non-temporal b128 stores.

#define KDIM     2048
#define KBATCH   16384
#define KN4      ((KBATCH * KDIM) / 4)   // 8,388,608 float4 granules
#define KCOLS4   (KDIM / 4)              // 512 float4 column-groups
#define BLOCK    256
#define DEPTH    16                       // async pipeline depth (tiles/block)
#define TILE4    (BLOCK * DEPTH)          // 4096 float4 = 64KB per block
#define GRID     (KN4 / TILE4)            // 2048 blocks, exact
#define STEP_B   (BLOCK * 16)             // 4096 bytes per step (imm offset)

typedef float float4v __attribute__((ext_vector_type(4)));
typedef int   v4i     __attribute__((ext_vector_type(4)));

#if defined(__AMDGCN__) && __has_builtin(__builtin_amdgcn_global_load_async_to_lds_b128)
#define USE_ASYNC_LDS 1
#else
#define USE_ASYNC_LDS 0
#endif

template <int N>
__device__ __forceinline__ void wait_asynccnt() {
#if defined(__AMDGCN__)
#if __has_builtin(__builtin_amdgcn_s_wait_asynccnt)
  __builtin_amdgcn_s_wait_asynccnt(N);
#else
  asm volatile("s_wait_asynccnt %0" ::"i"(N) : "memory");
#endif
  asm volatile("" ::: "memory");  // LDS consume must not hoist above the wait
#endif
}

#if USE_ASYNC_LDS
// Fill: one global base + one LDS base; the per-step 4KB stride goes into
// the instruction's IOFFSET, which the HW adds to BOTH addresses
// (CDNA5 ISA).
template <int IT>
__device__ __forceinline__ void fill_steps(const float4v* gbase, float4v* lbase) {
  if constexpr (IT < DEPTH) {
    __builtin_amdgcn_global_load_async_to_lds_b128((v4i*)gbase, (v4i*)lbase,
                                                   /*ioffset=*/IT * STEP_B,
                                                   /*cpol=*/0);
    fill_steps<IT + 1>(gbase, lbase);
  }
}

// Drain in completion order with compile-time ASYNCcnt thresholds.
template <int IT>
__device__ __forceinline__ void drain_steps(const float4v* xs, int tid, int i4,
                                            float4v wv0, float4v bv0,
                                            float4v wv1, float4v bv1,
                                            float4v* __restrict__ o4) {
  if constexpr (IT < DEPTH) {
    wait_asynccnt<DEPTH - 1 - IT>();           // tile IT has landed in LDS
    float4v xv = xs[IT * BLOCK + tid];         // ds_load_b128 base+imm
    float4v r  = (IT & 1) ? (xv * wv1 + bv1) : (xv * wv0 + bv0);
    __builtin_nontemporal_store(r, o4 + i4 + IT * BLOCK);  // NT b128 store
    drain_steps<IT + 1>(xs, tid, i4, wv0, bv0, wv1, bv1, o4);
  }
}
#endif

__global__ __launch_bounds__(BLOCK) void diag_linear_kernel(
    const float* __restrict__ x, const float* __restrict__ w,
    const float* __restrict__ b, float* __restrict__ out) {
  const int tid = (int)threadIdx.x;
  // Block-contiguous tile: per-step offset is 4KB, so every access is
  // base + imm -- no per-step address VALU at all.
  const int i4 = (int)blockIdx.x * TILE4 + tid;

  // TILE4 % KCOLS4 == 0 => column group alternates tid / tid+256 by step
  // parity; both scale/shift pairs live in registers (w/b are L2-resident).
  const float4v* __restrict__ w4 = (const float4v*)w;
  const float4v* __restrict__ b4 = (const float4v*)b;
  const float4v wv0 = w4[tid],         bv0 = b4[tid];
  const float4v wv1 = w4[tid + BLOCK], bv1 = b4[tid + BLOCK];

  const float4v* __restrict__ x4 = (const float4v*)x;
  float4v* __restrict__ o4 = (float4v*)out;

#if USE_ASYNC_LDS
  __shared__ float4v xs[DEPTH * BLOCK];  // 64KB (5 blocks = 40 waves / WGP)

  // 16 in-flight async b128 DMAs per lane (ASYNCcnt max is 63).
  fill_steps<0>(x4 + i4, &xs[tid]);
  drain_steps<0>(xs, tid, i4, wv0, bv0, wv1, bv1, o4);
#else
  // Fallback: direct non-temporal b128 stream on the LOADcnt scoreboard.
#pragma unroll
  for (int it = 0; it < DEPTH; ++it) {
    float4v xv = __builtin_nontemporal_load(x4 + i4 + it * BLOCK);
    float4v r  = (it & 1) ? (xv * wv1 + bv1) : (xv * wv0 + bv0);
    __builtin_nontemporal_store(r, o4 + i4 + it * BLOCK);
  }
#endif
}

extern "C" void kernel_launch(void* const* d_in, const int* in_sizes, int n_in,
                              void* d_out, int out_size, void* d_ws, size_t ws_size,
                              hipStream_t stream) {
  (void)in_sizes; (void)n_in; (void)out_size; (void)d_ws; (void)ws_size;
  const float* x = (const float*)d_in[0];
  const float* w = (const float*)d_in[1];
  const float* b = (const float*)d_in[2];
  float* out = (float*)d_out;
  diag_linear_kernel<<<dim3(GRID), dim3(BLOCK), 0, stream>>>(x, w, b, out);
}